// TransitionUp_90134183674397
// MI455X (gfx1250) — compile-verified
//
#include <hip/hip_runtime.h>

typedef __attribute__((ext_vector_type(2))) float v2f;
typedef __attribute__((ext_vector_type(8))) float v8f;

#define N_PTS   32768
#define M_PTS   8192
#define C_IN    256
#define C_OUT   128
#define CHPTS   4096   // LDS chunk of db points (48KB)

// -------------------------------------------------------------------------
// Kernel 1: h_sub = relu(x_sub @ W_sub + b_sub)   [8192,256]x[256,128]
// One wave = one 16x16 output tile, K stepped by 4 via V_WMMA_F32_16X16X4_F32
// -------------------------------------------------------------------------
__global__ __launch_bounds__(256) void k_gemm_hsub(const float* __restrict__ xs,
                                                   const float* __restrict__ Ws,
                                                   const float* __restrict__ bs,
                                                   float* __restrict__ h) {
  const int lane = threadIdx.x & 31;
  const int wv   = threadIdx.x >> 5;       // 0..7 -> column tile
  const int rowBase = blockIdx.x * 16;     // 512 blocks
  const int colBase = wv * 16;
  const int hi  = lane >> 4;               // 0: K pair {0,1}, 1: K pair {2,3}
  const int l16 = lane & 15;

  const int aRow = rowBase + l16;          // A: one matrix row per lane
  const int bCol = colBase + l16;          // B/C/D: one matrix col per lane

  v8f acc = {};
  for (int k = 0; k < C_IN; k += 4) {
    const int k0 = k + hi * 2;
    v2f a, b;
    a.x = xs[aRow * C_IN + k0];
    a.y = xs[aRow * C_IN + k0 + 1];
    b.x = Ws[k0 * C_OUT + bCol];
    b.y = Ws[(k0 + 1) * C_OUT + bCol];
    acc = __builtin_amdgcn_wmma_f32_16x16x4_f32(false, a, false, b,
                                                (short)0, acc, false, false);
  }
  const float bias = bs[bCol];
#pragma unroll
  for (int r = 0; r < 8; ++r) {
    const int row = rowBase + r + hi * 8;
    const float v = acc[r] + bias;
    h[row * C_OUT + bCol] = v > 0.f ? v : 0.f;
  }
}

// -------------------------------------------------------------------------
// Kernel 2: brute-force 3-NN via WMMA distance tiles.
// S(m,n) = |p_m|^2 - 2 q_n . p_m   (per-query constant |q|^2 added later)
// A-row (db point): (-2px, -2py, -2pz, |p|^2) ; B-col (query): (qx,qy,qz,1)
// Lane owns query (lane&15); lanes l and l^16 cover complementary db rows.
// -------------------------------------------------------------------------
#define INSERT3(v, gi)                                            \
  do {                                                            \
    if ((v) < s2) {                                               \
      if ((v) < s1) {                                             \
        if ((v) < s0) { s2=s1; i2=i1; s1=s0; i1=i0; s0=(v); i0=(gi); } \
        else          { s2=s1; i2=i1; s1=(v); i1=(gi); }          \
      } else          { s2=(v); i2=(gi); }                        \
    }                                                             \
  } while (0)

__global__ __launch_bounds__(256) void k_knn3(const float* __restrict__ pos,
                                              const float* __restrict__ psub,
                                              int*   __restrict__ oidx,
                                              float* __restrict__ ow) {
  __shared__ float sX[CHPTS], sY[CHPTS], sZ[CHPTS];
  const int tid  = threadIdx.x;
  const int lane = tid & 31;
  const int wv   = tid >> 5;
  const int l16  = lane & 15;
  const int hi   = lane >> 4;
  const int q    = (blockIdx.x * 8 + wv) * 16 + l16;  // this lane's query

  const float qx = pos[q * 3 + 0];
  const float qy = pos[q * 3 + 1];
  const float qz = pos[q * 3 + 2];
  const float qn2 = qx * qx + qy * qy + qz * qz;

  // Loop-invariant B operand: lanes 0-15 give K0,K1 = (qx,qy); 16-31 give K2,K3 = (qz,1)
  v2f b;
  b.x = hi ? qz : qx;
  b.y = hi ? 1.0f : qy;

  float s0 = 3.4e38f, s1 = 3.4e38f, s2 = 3.4e38f;
  int   i0 = 0, i1 = 0, i2 = 0;

  for (int base = 0; base < M_PTS; base += CHPTS) {
    __syncthreads();
    for (int t = tid; t < CHPTS * 3; t += 256) {      // uniform trip count
      const float v = psub[base * 3 + t];
      const int p = t / 3, c = t - p * 3;
      if (c == 0) sX[p] = v; else if (c == 1) sY[p] = v; else sZ[p] = v;
    }
    __syncthreads();

    for (int mb = 0; mb < CHPTS; mb += 16) {
      const int j = mb + l16;
      const float px = sX[j], py = sY[j], pz = sZ[j];
      const float pn2 = px * px + py * py + pz * pz;
      v2f a;
      a.x = hi ? (-2.0f * pz) : (-2.0f * px);
      a.y = hi ? pn2 : (-2.0f * py);
      v8f c = {};
      c = __builtin_amdgcn_wmma_f32_16x16x4_f32(false, a, false, b,
                                                (short)0, c, false, false);
#pragma unroll
      for (int r = 0; r < 8; ++r) {
        const float v = c[r];
        const int gi = base + mb + hi * 8 + r;        // db row of this value
        INSERT3(v, gi);
      }
    }
  }

  // Merge with partner lane (other half of db rows): snapshot first, then insert.
  const float t0 = __shfl_xor(s0, 16, 32);
  const float t1 = __shfl_xor(s1, 16, 32);
  const float t2 = __shfl_xor(s2, 16, 32);
  const int   j0 = __shfl_xor(i0, 16, 32);
  const int   j1 = __shfl_xor(i1, 16, 32);
  const int   j2 = __shfl_xor(i2, 16, 32);
  INSERT3(t0, j0);
  INSERT3(t1, j1);
  INSERT3(t2, j2);

  if (hi == 0) {
    float d0 = fmaxf(s0 + qn2, 0.0f);
    float d1 = fmaxf(s1 + qn2, 0.0f);
    float d2 = fmaxf(s2 + qn2, 0.0f);
    const float w0 = 1.0f / (d0 + 1e-16f);
    const float w1 = 1.0f / (d1 + 1e-16f);
    const float w2 = 1.0f / (d2 + 1e-16f);
    const float inv = 1.0f / (w0 + w1 + w2);
    oidx[q * 3 + 0] = i0; ow[q * 3 + 0] = w0 * inv;
    oidx[q * 3 + 1] = i1; ow[q * 3 + 1] = w1 * inv;
    oidx[q * 3 + 2] = i2; ow[q * 3 + 2] = w2 * inv;
  }
}

// -------------------------------------------------------------------------
// Kernel 3: out = relu(x @ W + b) + sum_k w_k * h_sub[idx_k]
// [32768,128]x[128,128] WMMA GEMM with fused gather-interp epilogue.
// -------------------------------------------------------------------------
__global__ __launch_bounds__(256) void k_gemm_out(const float* __restrict__ x,
                                                  const float* __restrict__ W,
                                                  const float* __restrict__ bb,
                                                  const float* __restrict__ h,
                                                  const int*   __restrict__ idx,
                                                  const float* __restrict__ wgt,
                                                  float* __restrict__ out) {
  const int lane = threadIdx.x & 31;
  const int wv   = threadIdx.x >> 5;
  const int rowBase = blockIdx.x * 16;   // 2048 blocks
  const int colBase = wv * 16;
  const int hi  = lane >> 4;
  const int l16 = lane & 15;

  const int aRow = rowBase + l16;
  const int bCol = colBase + l16;

  v8f acc = {};
  for (int k = 0; k < C_OUT; k += 4) {
    const int k0 = k + hi * 2;
    v2f a, b;
    a.x = x[aRow * C_OUT + k0];
    a.y = x[aRow * C_OUT + k0 + 1];
    b.x = W[k0 * C_OUT + bCol];
    b.y = W[(k0 + 1) * C_OUT + bCol];
    acc = __builtin_amdgcn_wmma_f32_16x16x4_f32(false, a, false, b,
                                                (short)0, acc, false, false);
  }
  const float bias = bb[bCol];
#pragma unroll
  for (int r = 0; r < 8; ++r) {
    const int row = rowBase + r + hi * 8;
    const int j0 = idx[row * 3 + 0], j1 = idx[row * 3 + 1], j2 = idx[row * 3 + 2];
    const float w0 = wgt[row * 3 + 0], w1 = wgt[row * 3 + 1], w2 = wgt[row * 3 + 2];
    float v = acc[r] + bias;
    v = v > 0.f ? v : 0.f;
    v += w0 * h[j0 * C_OUT + bCol]
       + w1 * h[j1 * C_OUT + bCol]
       + w2 * h[j2 * C_OUT + bCol];
    out[row * C_OUT + bCol] = v;
  }
}

// -------------------------------------------------------------------------
extern "C" void kernel_launch(void* const* d_in, const int* in_sizes, int n_in,
                              void* d_out, int out_size, void* d_ws, size_t ws_size,
                              hipStream_t stream) {
  const float* x     = (const float*)d_in[0];  // [32768,128]
  const float* xsub  = (const float*)d_in[1];  // [8192,256]
  const float* pos   = (const float*)d_in[2];  // [32768,3]
  const float* psub  = (const float*)d_in[3];  // [8192,3]
  const float* Wsub  = (const float*)d_in[4];  // [256,128]
  const float* bsub  = (const float*)d_in[5];  // [128]
  const float* W     = (const float*)d_in[6];  // [128,128]
  const float* b     = (const float*)d_in[7];  // [128]
  float* out = (float*)d_out;

  char* ws = (char*)d_ws;
  float* h   = (float*)ws;                                     // 8192*128 f32 = 4MB
  int*   idx = (int*)(ws + (size_t)M_PTS * C_OUT * 4);         // 32768*3 i32
  float* wgt = (float*)(ws + (size_t)M_PTS * C_OUT * 4
                           + (size_t)N_PTS * 3 * 4);           // 32768*3 f32

  k_gemm_hsub<<<M_PTS / 16, 256, 0, stream>>>(xsub, Wsub, bsub, h);
  k_knn3<<<N_PTS / (8 * 16), 256, 0, stream>>>(pos, psub, idx, wgt);
  k_gemm_out<<<N_PTS / 16, 256, 0, stream>>>(x, W, b, h, idx, wgt, out);
}